// VLLMKVCache_7103875908079
// MI455X (gfx1250) — compile-verified
//
#include <hip/hip_runtime.h>
#include <cstdint>
#include <cstddef>

// ---------------------------------------------------------------------------
// VLLM KV-cache scatter:  out = cache;  out[slot_mapping[i]] = input[i]
// Pure HBM-bandwidth problem: ~2.06 GiB moved -> ~90us floor @ 23.3 TB/s.
// ---------------------------------------------------------------------------

// Raw clang vector types (HIP_vector_type<> is a class and is rejected by
// __builtin_nontemporal_* and the amdgcn async builtins).
typedef float v4f __attribute__((ext_vector_type(4)));
typedef int   v4i __attribute__((vector_size(16)));

// gfx1250 async global<->LDS path (ASYNCcnt-tracked), guarded so the file
// compiles on toolchains that don't expose the builtins.
#if defined(__gfx1250__) && \
    __has_builtin(__builtin_amdgcn_global_load_async_to_lds_b128) && \
    __has_builtin(__builtin_amdgcn_global_store_async_from_lds_b128)
#define USE_ASYNC_LDS 1
#else
#define USE_ASYNC_LDS 0
#endif

#if defined(__gfx1250__)
#if __has_builtin(__builtin_amdgcn_s_wait_asynccnt)
#define WAIT_ASYNC0() __builtin_amdgcn_s_wait_asynccnt(0)
#else
#define WAIT_ASYNC0() asm volatile("s_wait_asynccnt 0" ::: "memory")
#endif
#else
#define WAIT_ASYNC0() ((void)0)
#endif

#if USE_ASYNC_LDS
// Builtin signature (confirmed by clang diagnostic): param 0 is
// 'int __attribute__((vector_size(16))) __device__ *'  ==  AS(1) v4i*.
typedef __attribute__((address_space(1))) v4i* gv4i_p;  // global v4i*
typedef __attribute__((address_space(3))) v4i* lv4i_p;  // LDS v4i*
#endif

// ---------------------------------------------------------------------------
// Kernel A (placed first so the disasm snippet shows the async ops):
// scatter input rows into out at slot_mapping positions.
// One block per token; 256 lanes x 16B = one 4 KiB row per pass.
// Data path: global --(global_load_async_to_lds_b128)--> LDS
//                  --(global_store_async_from_lds_b128)--> global,
// synchronized wave-wide with s_wait_asynccnt (ASYNCcnt).
// ---------------------------------------------------------------------------
__global__ __launch_bounds__(256) void kv_scatter_kernel(
    const v4i* __restrict__ src, const long long* __restrict__ slots,
    v4i* __restrict__ dst, int row4) {
  const int token = blockIdx.x;
  const long long slot = slots[token];
  const v4i* s = src + (size_t)token * (size_t)row4;
  v4i*       d = dst + (size_t)slot  * (size_t)row4;

#if USE_ASYNC_LDS
  __shared__ v4i buf[256];
  for (int c = threadIdx.x; c < row4; c += blockDim.x) {
    __builtin_amdgcn_global_load_async_to_lds_b128(
        (gv4i_p)(s + c), (lv4i_p)&buf[threadIdx.x], 0, 0);
    WAIT_ASYNC0();
    __builtin_amdgcn_global_store_async_from_lds_b128(
        (gv4i_p)(d + c), (lv4i_p)&buf[threadIdx.x], 0, 0);
    WAIT_ASYNC0();
  }
#else
  for (int c = threadIdx.x; c < row4; c += blockDim.x) {
    v4i v = __builtin_nontemporal_load(s + c);
    __builtin_nontemporal_store(v, d + c);
  }
#endif
}

// ---------------------------------------------------------------------------
// Kernel B: bulk copy cache -> out (1 GiB, never re-read => TH=NT both ways).
// Per-block contiguous tiles of 4*blockDim v4f: thread t's four accesses are
// base+t, +256, +512, +768 -> constant 4096/8192/12288-byte offsets that fold
// into the b128 instruction offset (one base VGPR pair, one add per iter),
// and each load/store is a fully contiguous 512 B/wave transaction.
// ---------------------------------------------------------------------------
__global__ __launch_bounds__(256) void cache_copy_kernel(
    const v4f* __restrict__ src, v4f* __restrict__ dst, size_t n4) {
  const size_t bdim    = blockDim.x;                 // 256
  const size_t tile    = bdim * 4;                   // 1024 v4f / block-iter
  const size_t gstride = (size_t)gridDim.x * tile;
  size_t base = (size_t)blockIdx.x * tile + threadIdx.x;

  for (; base + 3 * bdim < n4; base += gstride) {
    // keep the read stream ahead of us in flight (global_prefetch_b8)
    __builtin_prefetch(src + base + gstride, 0, 1);
    v4f a = __builtin_nontemporal_load(src + base);
    v4f b = __builtin_nontemporal_load(src + base + bdim);
    v4f c = __builtin_nontemporal_load(src + base + 2 * bdim);
    v4f d = __builtin_nontemporal_load(src + base + 3 * bdim);
    __builtin_nontemporal_store(a, dst + base);
    __builtin_nontemporal_store(b, dst + base + bdim);
    __builtin_nontemporal_store(c, dst + base + 2 * bdim);
    __builtin_nontemporal_store(d, dst + base + 3 * bdim);
  }
  // final partial tile (unused for our exact sizes, kept for generality)
  for (size_t i = base; i < n4; i += bdim) {
    v4f v = __builtin_nontemporal_load(src + i);
    __builtin_nontemporal_store(v, dst + i);
  }
}

// ---------------------------------------------------------------------------
// Host launcher. Inputs (reference order): input f32[8192,8,128],
// cache f32[262144,8,128], slot_mapping i64[8192]. Output: full cache.
// ---------------------------------------------------------------------------
extern "C" void kernel_launch(void* const* d_in, const int* in_sizes, int n_in,
                              void* d_out, int out_size, void* d_ws, size_t ws_size,
                              hipStream_t stream) {
  const v4f*       inp   = (const v4f*)d_in[0];
  const v4f*       cache = (const v4f*)d_in[1];
  const long long* slots = (const long long*)d_in[2];  // int64 per reference
  v4f*             out   = (v4f*)d_out;

  // 1) out = cache  (1 GiB, NT b128 streaming copy; 8192 blocks -> 64K waves)
  const size_t n4 = (size_t)out_size / 4;
  cache_copy_kernel<<<8192, 256, 0, stream>>>(cache, out, n4);

  // 2) out[slots[i]] = input[i]  (stream-ordered after the copy)
  const int ntok = in_sizes[2];                 // 8192 tokens
  const int row4 = (in_sizes[0] / ntok) / 4;    // 1024 floats -> 256 v4f
  kv_scatter_kernel<<<ntok, 256, 0, stream>>>(
      (const v4i*)inp, slots, (v4i*)out, row4);
}